// GIN_57870389346580
// MI455X (gfx1250) — compile-verified
//
#include <hip/hip_runtime.h>
#include <hip/hip_bf16.h>

// GIN (3 layers) for MI455X / gfx1250, wave32.
// Memory-bound workload: scatter-add dominates; GEMMs use V_WMMA_F32_16X16X4_F32.

typedef float v2f __attribute__((ext_vector_type(2)));
typedef float v8f __attribute__((ext_vector_type(8)));

// ---------------- zero workspace ----------------
__global__ void gin_zero_f4(float4* __restrict__ p, long long n4) {
    long long i = (long long)blockIdx.x * blockDim.x + threadIdx.x;
    if (i < n4) p[i] = make_float4(0.f, 0.f, 0.f, 0.f);
}

// ---------------- edge scatter-sum: agg[dst] += x[src] ----------------
// One (edge, feature-pair) per thread: 32 lanes cover F=64 as float2.
__global__ void gin_scatter(const float* __restrict__ x,
                            const long long* __restrict__ ei,
                            float* __restrict__ agg, long long E) {
    long long idx = (long long)blockIdx.x * blockDim.x + threadIdx.x;
    long long e = idx >> 5;
    if (e >= E) return;
    int f = (int)(idx & 31) * 2;
    long long s = ei[e];        // src row
    long long d = ei[E + e];    // dst row
    v2f v = *(const v2f*)(x + s * 64 + f);
    float* dst = agg + d * 64 + f;
    __hip_atomic_fetch_add(dst,     v.x, __ATOMIC_RELAXED, __HIP_MEMORY_SCOPE_AGENT);
    __hip_atomic_fetch_add(dst + 1, v.y, __ATOMIC_RELAXED, __HIP_MEMORY_SCOPE_AGENT);
}

// ---------------- fused MLP: out = relu((x+agg)@w1 + b1) @ w2 + b2 ----------------
// Block = 128 threads = 4 waves; each wave computes a 16x64 output tile
// via V_WMMA_F32_16X16X4_F32 (16 K-steps x 4 N-tiles per GEMM).
__global__ __launch_bounds__(128) void gin_mlp(
    const float* __restrict__ x, const float* __restrict__ agg,
    const float* __restrict__ w1, const float* __restrict__ b1,
    const float* __restrict__ w2, const float* __restrict__ b2,
    float* __restrict__ out, int N)
{
    __shared__ float sW1[64 * 64];
    __shared__ float sW2[64 * 64];
    __shared__ float sB1[64];
    __shared__ float sB2[64];
    __shared__ float sStage[4][16 * 64];   // per-wave C->A relayout tile

    const int tid = threadIdx.x;

    // cooperative weight/bias staging into LDS (float4 wide)
    {
        const float4* w1v = (const float4*)w1;
        const float4* w2v = (const float4*)w2;
        float4* s1 = (float4*)sW1;
        float4* s2 = (float4*)sW2;
#pragma unroll
        for (int i = 0; i < 8; ++i) {
            s1[tid + i * 128] = w1v[tid + i * 128];
            s2[tid + i * 128] = w2v[tid + i * 128];
        }
        if (tid < 64) { sB1[tid] = b1[tid]; sB2[tid] = b2[tid]; }
    }
    __syncthreads();

    const int lane = tid & 31;
    const int wave = tid >> 5;
    const int nl   = lane & 15;     // N index within 16 / A-matrix row
    const int hi   = lane >> 4;     // half-wave select
    const int koff = hi * 2;        // K offset for A/B fragments (32-bit layout)

    const int tileRow = (blockIdx.x * 4 + wave) * 16;
    const int arow    = tileRow + nl;
    const bool rowOK  = arow < N;
    const size_t abase = (size_t)arow * 64;

    // ---- load A fragments for GEMM1: h = x + agg  (A layout: lane=row, koff selects K pair)
    v2f afrag[16];
#pragma unroll
    for (int ks = 0; ks < 16; ++ks) {
        const int c = ks * 4 + koff;
        v2f xv = {0.f, 0.f}, av = {0.f, 0.f};
        if (rowOK) {
            xv = *(const v2f*)(x + abase + c);
            av = *(const v2f*)(agg + abase + c);
        }
        afrag[ks].x = xv.x + av.x;
        afrag[ks].y = xv.y + av.y;
    }

    const v8f vzero = {0.f, 0.f, 0.f, 0.f, 0.f, 0.f, 0.f, 0.f};

    // ---- GEMM1: (x+agg) @ w1
    v8f acc[4] = {vzero, vzero, vzero, vzero};
#pragma unroll
    for (int ks = 0; ks < 16; ++ks) {
        const int kb = ks * 4 + koff;
#pragma unroll
        for (int nt = 0; nt < 4; ++nt) {
            v2f bfrag;
            bfrag.x = sW1[kb * 64 + nt * 16 + nl];
            bfrag.y = sW1[(kb + 1) * 64 + nt * 16 + nl];
            acc[nt] = __builtin_amdgcn_wmma_f32_16x16x4_f32(
                false, afrag[ks], false, bfrag, (short)0, acc[nt], false, false);
        }
    }

    // ---- bias + ReLU, stage C-layout result into LDS for A-layout reload
#pragma unroll
    for (int nt = 0; nt < 4; ++nt) {
        const float bb = sB1[nt * 16 + nl];
#pragma unroll
        for (int r = 0; r < 8; ++r) {
            float v = acc[nt][r] + bb;            // C layout: M = r + hi*8, N = nt*16 + nl
            sStage[wave][(r + hi * 8) * 64 + nt * 16 + nl] = v > 0.f ? v : 0.f;
        }
    }
    __syncthreads();

    // ---- reload h1 as A fragments
    v2f a2[16];
#pragma unroll
    for (int ks = 0; ks < 16; ++ks) {
        a2[ks] = *(const v2f*)&sStage[wave][nl * 64 + ks * 4 + koff];
    }

    // ---- GEMM2: h1 @ w2
    v8f acc2[4] = {vzero, vzero, vzero, vzero};
#pragma unroll
    for (int ks = 0; ks < 16; ++ks) {
        const int kb = ks * 4 + koff;
#pragma unroll
        for (int nt = 0; nt < 4; ++nt) {
            v2f bfrag;
            bfrag.x = sW2[kb * 64 + nt * 16 + nl];
            bfrag.y = sW2[(kb + 1) * 64 + nt * 16 + nl];
            acc2[nt] = __builtin_amdgcn_wmma_f32_16x16x4_f32(
                false, a2[ks], false, bfrag, (short)0, acc2[nt], false, false);
        }
    }

    // ---- bias + store
#pragma unroll
    for (int nt = 0; nt < 4; ++nt) {
        const float bb = sB2[nt * 16 + nl];
#pragma unroll
        for (int r = 0; r < 8; ++r) {
            const int orow = tileRow + r + hi * 8;
            if (orow < N)
                out[(size_t)orow * 64 + nt * 16 + nl] = acc2[nt][r] + bb;
        }
    }
}

extern "C" void kernel_launch(void* const* d_in, const int* in_sizes, int n_in,
                              void* d_out, int out_size, void* d_ws, size_t ws_size,
                              hipStream_t stream) {
    const float* x        = (const float*)d_in[0];
    const long long* ei   = (const long long*)d_in[1];   // int64 edge_index [2, E]
    const int N = in_sizes[0] / 64;
    const long long E = in_sizes[1] / 2;

    // workspace: agg | hA | hB  (each N*64 floats)
    float* agg = (float*)d_ws;
    float* hA  = agg + (size_t)N * 64;
    float* hB  = hA  + (size_t)N * 64;

    const float* in = x;
    for (int l = 0; l < 3; ++l) {
        const float* w1 = (const float*)d_in[2 + 4 * l];
        const float* b1 = (const float*)d_in[3 + 4 * l];
        const float* w2 = (const float*)d_in[4 + 4 * l];
        const float* b2 = (const float*)d_in[5 + 4 * l];
        float* out = (l == 2) ? (float*)d_out : ((l == 0) ? hA : hB);

        const long long n4 = (long long)N * 16;  // N*64/4 float4s
        gin_zero_f4<<<(unsigned)((n4 + 255) / 256), 256, 0, stream>>>((float4*)agg, n4);

        const long long sthreads = E * 32;
        gin_scatter<<<(unsigned)((sthreads + 255) / 256), 256, 0, stream>>>(in, ei, agg, E);

        gin_mlp<<<(unsigned)((N + 63) / 64), 128, 0, stream>>>(
            in, agg, w1, b1, w2, b2, out, N);

        in = out;
    }
}